// KVDequantizer_18923625906733
// MI455X (gfx1250) — compile-verified
//
#include <hip/hip_runtime.h>

// Q4 KV-cache dequantization for MI455X (gfx1250, wave32).
//
// Layout facts (from the reference):
//   packed : int32[B * N_BLOCKS * 16], each holds a byte (two nibbles)
//   scale  : float[B * N_BLOCKS]   (one per 32-float quant block)
//   bias   : float[B * N_BLOCKS]
//   out    : float[B * N_BLOCKS * 32], flat order == input order
//            (the head-dim reshapes in the reference are pure views)
//
// Each thread consumes 4 consecutive packed int32 (one global_load_b128,
// 16 B) and produces 8 floats (two global_store_b128, 32 B). Since the
// thread's packed base index is 4-aligned and a quant block is 16 packed
// ints, all 4 ints share one scale/bias -> exactly one b32 load of each.
// Streams are single-touch, so use non-temporal (TH=NT) loads/stores to
// keep the ~100 MiB stream out of L2's way; scale/bias stay RT because
// 4 adjacent lanes reuse each value through the cache.

typedef __attribute__((ext_vector_type(4))) int   v4i;
typedef __attribute__((ext_vector_type(4))) float v4f;

__global__ __launch_bounds__(256) void kv_dequant_q4_kernel(
    const int*   __restrict__ packed,   // [n_packed] int32, each in [0,256)
    const float* __restrict__ scale,    // [n_packed/16]
    const float* __restrict__ bias,     // [n_packed/16]
    float*       __restrict__ out,      // [n_packed*2]
    int n_threads)                      // = n_packed / 4
{
  const int t = blockIdx.x * blockDim.x + threadIdx.x;
  if (t >= n_threads) return;

  // 16 B coalesced non-temporal load: lane i reads bytes [16i, 16i+16).
  const v4i p = __builtin_nontemporal_load((const v4i*)packed + t);

  // Quant-block id: (4*t) / 16 == t >> 2. One scale + one bias per thread.
  const int   blk = t >> 2;
  const float s   = scale[blk];
  const float b   = bias[blk];

  // low nibble first, then high nibble (matches jnp.stack([low,high],axis=3)).
  v4f o0, o1;
  o0.x = __builtin_fmaf((float)( p.x        & 15), s, b);
  o0.y = __builtin_fmaf((float)((p.x >> 4)  & 15), s, b);
  o0.z = __builtin_fmaf((float)( p.y        & 15), s, b);
  o0.w = __builtin_fmaf((float)((p.y >> 4)  & 15), s, b);
  o1.x = __builtin_fmaf((float)( p.z        & 15), s, b);
  o1.y = __builtin_fmaf((float)((p.z >> 4)  & 15), s, b);
  o1.z = __builtin_fmaf((float)( p.w        & 15), s, b);
  o1.w = __builtin_fmaf((float)((p.w >> 4)  & 15), s, b);

  // Two 16 B non-temporal stores: lane i writes bytes [32i, 32i+32).
  v4f* dst = (v4f*)out + 2 * t;
  __builtin_nontemporal_store(o0, dst);
  __builtin_nontemporal_store(o1, dst + 1);
}

extern "C" void kernel_launch(void* const* d_in, const int* in_sizes, int n_in,
                              void* d_out, int out_size, void* d_ws, size_t ws_size,
                              hipStream_t stream) {
  const int*   k_packed = (const int*)  d_in[0];
  const float* k_scale  = (const float*)d_in[1];
  const float* k_bias   = (const float*)d_in[2];
  const int*   v_packed = (const int*)  d_in[3];
  const float* v_scale  = (const float*)d_in[4];
  const float* v_bias   = (const float*)d_in[5];
  float* out = (float*)d_out;

  const int n_packed      = in_sizes[0];      // B * N_BLOCKS * 16 = 4,194,304
  const int n_threads     = n_packed / 4;     // 1,048,576
  const int floats_per_t  = n_packed * 2;     // 8,388,608 floats per tensor

  const int block = 256;                      // 8 wave32 waves
  const int grid  = (n_threads + block - 1) / block;   // 4096

  // K then V, concatenated flat in d_out (tuple return order).
  kv_dequant_q4_kernel<<<grid, block, 0, stream>>>(
      k_packed, k_scale, k_bias, out, n_threads);
  kv_dequant_q4_kernel<<<grid, block, 0, stream>>>(
      v_packed, v_scale, v_bias, out + floats_per_t, n_threads);
}